// DownsampleModel_20564303414104
// MI455X (gfx1250) — compile-verified
//
#include <hip/hip_runtime.h>

typedef float v2f __attribute__((ext_vector_type(2)));
typedef float v4f __attribute__((ext_vector_type(4)));
typedef float v8f __attribute__((ext_vector_type(8)));

// ---------------------------------------------------------------------------
// Rulebook inversion: for each offset k, inv[k][rout[k,p]] = rin[k,p].
// Within one offset every output row appears at most once, so plain stores
// are conflict-free.  inv is pre-filled with -1 (memset 0xFF).
// Pad entries have rout == n_out and are skipped.
// ---------------------------------------------------------------------------
__global__ void build_inv(const int* __restrict__ rin, const int* __restrict__ rout,
                          int* __restrict__ inv, int P, int n_out)
{
    const size_t total  = (size_t)27 * P;
    const size_t stride = (size_t)gridDim.x * blockDim.x;
    for (size_t e = (size_t)blockIdx.x * blockDim.x + threadIdx.x; e < total; e += stride) {
        const int o = rout[e];
        if (o < n_out) {
            const size_t k = e / (size_t)P;
            inv[k * (size_t)n_out + o] = rin[e];
        }
    }
}

// ---------------------------------------------------------------------------
// Output-stationary sparse conv:
//   out[o] = sum_k feats[inv[k][o]] @ W[k]        (inv[k][o] < 0 -> skip)
// Grid: (ceil(n_out/128), COUT/64).  Block: 256 threads = 8 waves (wave32).
// Each wave owns a 16-row x 64-col fp32 accumulator tile held in VGPRs across
// all 27 offsets; per offset it gathers 16 input rows and runs
// V_WMMA_F32_16X16X4_F32 over CIN/4 K-steps x 4 N-tiles.  No atomics; a single
// coalesced store at the end (so outputs need no pre-zeroing).
// ---------------------------------------------------------------------------
template<int CIN>
__global__ __launch_bounds__(256)
void sparse_conv_os(const float* __restrict__ feats,  // [n_in, CIN]
                    const float* __restrict__ W,      // [27, CIN, COUT]
                    const int*   __restrict__ inv,    // [27, n_out]
                    float*       __restrict__ out,    // [n_out, COUT]
                    int n_out, int COUT)
{
    __shared__ float sW[CIN * 64];   // W[k][:, ct*64 : ct*64+64]

    const int ct    = blockIdx.y;    // 64-wide column tile
    const int tid   = threadIdx.x;
    const int wave  = tid >> 5;
    const int lane  = tid & 31;
    const int l15   = lane & 15;
    const int khalf = lane >> 4;     // 0: K pair {0,1}; 1: K pair {2,3}

    const int rbase = blockIdx.x * 128 + wave * 16;
    const int orow  = rbase + l15;
    const bool rvalid = (orow < n_out);

    v8f acc[4];
    #pragma unroll
    for (int nt = 0; nt < 4; ++nt) {
        v8f z = {0.f, 0.f, 0.f, 0.f, 0.f, 0.f, 0.f, 0.f};
        acc[nt] = z;
    }

    const int c4stride = COUT >> 2;

    for (int k = 0; k < 27; ++k) {
        __syncthreads();
        // ---- stage this offset's weight tile into LDS (coalesced float4) ----
        {
            const v4f* Wk  = (const v4f*)(W + (size_t)k * CIN * COUT);
            v4f* sW4 = (v4f*)sW;
            for (int e = tid; e < CIN * 16; e += 256) {
                const int row = e >> 4;
                const int c4  = e & 15;
                sW4[e] = Wk[row * c4stride + ct * 16 + c4];
            }
        }
        // ---- prefetch next offset's weight tile (global_prefetch_b8) ----
        if (k < 26 && tid < 2 * CIN) {
            const float* Wn = W + (size_t)(k + 1) * CIN * COUT;
            const int row = tid >> 1;
            const int h   = tid & 1;
            __builtin_prefetch(Wn + (size_t)row * COUT + ct * 64 + h * 32, 0, 1);
        }
        __syncthreads();

        // ---- gather this offset's 16 input rows into registers ----
        const int idx = rvalid ? inv[(size_t)k * n_out + orow] : -1;
        const bool valid = (idx >= 0);
        const v2f* arow = (const v2f*)(feats + (size_t)(valid ? idx : 0) * CIN);

        v2f afrag[CIN / 4];
        #pragma unroll
        for (int kk = 0; kk < CIN / 4; ++kk) {
            // A 16x4 fp32 layout: lanes 0-15 hold K=kk*4+{0,1}, lanes 16-31 K=kk*4+{2,3}
            v2f a = arow[kk * 2 + khalf];
            afrag[kk].x = valid ? a.x : 0.0f;
            afrag[kk].y = valid ? a.y : 0.0f;
        }

        // ---- WMMA accumulate: D(16x16) += A(16x4) x B(4x16), 4 N-tiles ----
        #pragma unroll
        for (int kk = 0; kk < CIN / 4; ++kk) {
            const v2f a = afrag[kk];
            const int kb = (kk * 4 + khalf * 2) * 64;
            #pragma unroll
            for (int nt = 0; nt < 4; ++nt) {
                v2f b;
                b.x = sW[kb +      nt * 16 + l15];
                b.y = sW[kb + 64 + nt * 16 + l15];
                acc[nt] = __builtin_amdgcn_wmma_f32_16x16x4_f32(
                    false, a, false, b, (short)0, acc[nt], false, false);
            }
        }
    }

    // ---- single coalesced store (C/D layout: vgpr i -> M = i + khalf*8) ----
    #pragma unroll
    for (int i = 0; i < 8; ++i) {
        const int row = rbase + i + khalf * 8;
        if (row < n_out) {
            float* orow_ptr = out + (size_t)row * COUT + ct * 64;
            #pragma unroll
            for (int nt = 0; nt < 4; ++nt) {
                orow_ptr[nt * 16 + l15] = acc[nt][i];
            }
        }
    }
}

// ---------------------------------------------------------------------------
// BatchNorm stats: per-channel sum and sum-of-squares (biased stats)
// blockDim.x == COUT; each block strides over rows; one atomicAdd per thread.
// ---------------------------------------------------------------------------
template<int COUT>
__global__ void bn_stats(const float* __restrict__ x, int n,
                         float* __restrict__ sums, float* __restrict__ sqs)
{
    const int c = threadIdx.x;
    float s = 0.0f, q = 0.0f;
    for (int r = blockIdx.x; r < n; r += gridDim.x) {
        const float v = x[(size_t)r * COUT + c];
        s += v;
        q += v * v;
    }
    atomicAdd(sums + c, s);
    atomicAdd(sqs  + c, q);
}

// ---------------------------------------------------------------------------
// In-place BN (training-mode, biased var) + ReLU
// ---------------------------------------------------------------------------
template<int COUT>
__global__ void bn_apply(float* __restrict__ x, int n,
                         const float* __restrict__ sums, const float* __restrict__ sqs,
                         const float* __restrict__ g, const float* __restrict__ b)
{
    const size_t total  = (size_t)n * COUT;
    const size_t stride = (size_t)gridDim.x * blockDim.x;
    const float  inv_n  = 1.0f / (float)n;
    for (size_t e = (size_t)blockIdx.x * blockDim.x + threadIdx.x; e < total; e += stride) {
        const int   c   = (int)(e % COUT);
        const float mu  = sums[c] * inv_n;
        const float var = sqs[c] * inv_n - mu * mu;
        const float y   = (x[e] - mu) * rsqrtf(var + 1e-5f) * g[c] + b[c];
        x[e] = fmaxf(y, 0.0f);
    }
}

// ---------------------------------------------------------------------------
extern "C" void kernel_launch(void* const* d_in, const int* in_sizes, int n_in_args,
                              void* d_out, int out_size, void* d_ws, size_t ws_size,
                              hipStream_t stream)
{
    (void)n_in_args; (void)ws_size; (void)out_size;

    const float* feats = (const float*)d_in[0];
    const float* W1 = (const float*)d_in[1];
    const float* W2 = (const float*)d_in[2];
    const float* W3 = (const float*)d_in[3];
    const float* g1 = (const float*)d_in[4];
    const float* b1 = (const float*)d_in[5];
    const float* g2 = (const float*)d_in[6];
    const float* b2 = (const float*)d_in[7];
    const float* g3 = (const float*)d_in[8];
    const float* b3 = (const float*)d_in[9];
    const int* rb1_in  = (const int*)d_in[10];
    const int* rb1_out = (const int*)d_in[11];
    const int* rb2_in  = (const int*)d_in[12];
    const int* rb2_out = (const int*)d_in[13];
    const int* rb3_in  = (const int*)d_in[14];
    const int* rb3_out = (const int*)d_in[15];

    const int C  = 64;
    const int N  = in_sizes[0] / C;          // 180000 occupied input voxels
    const int M  = 32 * 32 * 32;             // stride-2 output voxels
    const int P1 = in_sizes[10] / 27;
    const int P2 = in_sizes[12] / 27;
    const int P3 = in_sizes[14] / 27;

    // ---- workspace layout ----
    char*  ws   = (char*)d_ws;
    size_t off  = 0;
    auto carve = [&](size_t bytes) { void* p = ws + off; off += (bytes + 255) & ~(size_t)255; return p; };

    float* x1   = (float*)carve((size_t)N * C * sizeof(float));          // [N,64]
    float* x2   = (float*)carve((size_t)M * 2 * C * sizeof(float));      // [M,128]
    int*   inv1 = (int*)  carve((size_t)27 * N * sizeof(int));           // [27,N]
    int*   inv2 = (int*)  carve((size_t)27 * M * sizeof(int));           // [27,M]
    int*   inv3 = (int*)  carve((size_t)27 * M * sizeof(int));           // [27,M]
    float* st   = (float*)carve(640 * sizeof(float));
    float* s1 = st;        float* q1 = st + 64;
    float* s2 = st + 128;  float* q2 = st + 256;
    float* s3 = st + 384;  float* q3 = st + 512;

    float* y3 = (float*)d_out;               // [M, 128]

    // ---- init: inverse maps to -1, stats to 0 (conv outputs need no zeroing) ----
    hipMemsetAsync(inv1, 0xFF, (size_t)27 * N * sizeof(int), stream);
    hipMemsetAsync(inv2, 0xFF, (size_t)27 * M * sizeof(int), stream);
    hipMemsetAsync(inv3, 0xFF, (size_t)27 * M * sizeof(int), stream);
    hipMemsetAsync(st, 0, 640 * sizeof(float), stream);

    // ---- invert rulebooks (conflict-free stores) ----
    build_inv<<<dim3(2048), dim3(256), 0, stream>>>(rb1_in, rb1_out, inv1, P1, N);
    build_inv<<<dim3(512),  dim3(256), 0, stream>>>(rb2_in, rb2_out, inv2, P2, M);
    build_inv<<<dim3(512),  dim3(256), 0, stream>>>(rb3_in, rb3_out, inv3, P3, M);

    // ---- Layer 1: submanifold 3x3x3, 64 -> 64, N outputs ----
    {
        dim3 grid((N + 127) / 128, 1);
        sparse_conv_os<64><<<grid, 256, 0, stream>>>(feats, W1, inv1, x1, N, 64);
        bn_stats<64><<<dim3(512), dim3(64), 0, stream>>>(x1, N, s1, q1);
        bn_apply<64><<<dim3(1024), dim3(256), 0, stream>>>(x1, N, s1, q1, g1, b1);
    }

    // ---- Layer 2: stride-2, 64 -> 128, M outputs ----
    {
        dim3 grid((M + 127) / 128, 2);
        sparse_conv_os<64><<<grid, 256, 0, stream>>>(x1, W2, inv2, x2, M, 128);
        bn_stats<128><<<dim3(512), dim3(128), 0, stream>>>(x2, M, s2, q2);
        bn_apply<128><<<dim3(1024), dim3(256), 0, stream>>>(x2, M, s2, q2, g2, b2);
    }

    // ---- Layer 3: submanifold 3x3x3, 128 -> 128, M outputs ----
    {
        dim3 grid((M + 127) / 128, 2);
        sparse_conv_os<128><<<grid, 256, 0, stream>>>(x2, W3, inv3, y3, M, 128);
        bn_stats<128><<<dim3(512), dim3(128), 0, stream>>>(y3, M, s3, q3);
        bn_apply<128><<<dim3(1024), dim3(256), 0, stream>>>(y3, M, s3, q3, g3, b3);
    }
}